// LossModul_16277926052544
// MI455X (gfx1250) — compile-verified
//
#include <hip/hip_runtime.h>
#include <hip/hip_bf16.h>

// ---------------------------------------------------------------------------
// YOLO-v1 style loss, reduced to a single f32 scalar.
// Memory-bound (96 MB in / 4 B out -> ~4.1 us floor @ 23.3 TB/s HBM).
// Kernel 1: grid-strided elementwise loss + deterministic block reduction.
// Kernel 2: single wave32; final cross-lane reduction via V_WMMA_F32_16X16X4_F32.
// ---------------------------------------------------------------------------

typedef float v2f __attribute__((ext_vector_type(2)));
typedef float v8f __attribute__((ext_vector_type(8)));

#define S_GRID 7
#define NCH 30
#define RATIO (1.0f / 7.0f)
#define EPSF 1e-6f
#define LAMBDA_COORD 5.0f
#define LAMBDA_NOOBJ 0.5f

__device__ __forceinline__ float iou_abs(float acx, float acy, float aw, float ah,
                                         float bcx, float bcy, float bw, float bh) {
    float ax1 = acx - 0.5f * aw, ax2 = acx + 0.5f * aw;
    float ay1 = acy - 0.5f * ah, ay2 = acy + 0.5f * ah;
    float bx1 = bcx - 0.5f * bw, bx2 = bcx + 0.5f * bw;
    float by1 = bcy - 0.5f * bh, by2 = bcy + 0.5f * bh;
    float iw = fmaxf(fminf(ax2, bx2) - fmaxf(ax1, bx1), 0.0f);
    float ih = fmaxf(fminf(ay2, by2) - fmaxf(ay1, by1), 0.0f);
    float inter  = iw * ih;
    float area_a = (ax2 - ax1) * (ay2 - ay1);
    float area_b = (bx2 - bx1) * (by2 - by1);
    return inter / (area_a + area_b - inter);   // reference has no eps here
}

__device__ __forceinline__ float cell_loss(const float* __restrict__ P,
                                           const float* __restrict__ T,
                                           float gx, float gy) {
    // cell base is 30 floats = 120 bytes -> always 8-byte aligned: float2 loads.
    const float2* P2 = reinterpret_cast<const float2*>(P);
    const float2* T2 = reinterpret_cast<const float2*>(T);
    float2 p01 = P2[0], p23 = P2[1], p45 = P2[2], p67 = P2[3], p89 = P2[4];
    float2 t01 = T2[0], t23 = T2[1], t45 = T2[2];
    float p0 = p01.x, p1 = p01.y, p2 = p23.x, p3 = p23.y, p4 = p45.x;
    float p5 = p45.y, p6 = p67.x, p7 = p67.y, p8 = p89.x, p9 = p89.y;
    float t0 = t01.x, t1 = t01.y, t2 = t23.x, t3 = t23.y, t4 = t45.x;

    bool obj = t4 > 0.0f;

    // absolute-coordinate centers (w,h already absolute)
    float a1cx = (p0 + gx) * RATIO, a1cy = (p1 + gy) * RATIO;
    float a2cx = (p5 + gx) * RATIO, a2cy = (p6 + gy) * RATIO;
    float atcx = (t0 + gx) * RATIO, atcy = (t1 + gy) * RATIO;

    float iou1 = iou_abs(a1cx, a1cy, p2, p3, atcx, atcy, t2, t3);
    float iou2 = iou_abs(a2cx, a2cy, p7, p8, atcx, atcy, t2, t3);
    bool resp = iou1 > iou2;

    float m1 = (obj && resp)  ? 1.0f : 0.0f;
    float m2 = (obj && !resp) ? 1.0f : 0.0f;
    float mo = obj ? 1.0f : 0.0f;
    float mn = obj ? 0.0f : 1.0f;

    float dx1 = p0 - t0, dy1 = p1 - t1;
    float dx2 = p5 - t0, dy2 = p6 - t1;
    float loss_xy = m1 * (dx1 * dx1 + dy1 * dy1) + m2 * (dx2 * dx2 + dy2 * dy2);

    float st2 = sqrtf(fmaxf(t2, EPSF));
    float st3 = sqrtf(fmaxf(t3, EPSF));
    float sw1 = sqrtf(fmaxf(p2, EPSF)) - st2;
    float sh1 = sqrtf(fmaxf(p3, EPSF)) - st3;
    float sw2 = sqrtf(fmaxf(p7, EPSF)) - st2;
    float sh2 = sqrtf(fmaxf(p8, EPSF)) - st3;
    float loss_wh = m1 * (sw1 * sw1 + sh1 * sh1) + m2 * (sw2 * sw2 + sh2 * sh2);

    float d1 = p4 - iou1, d2 = p9 - iou2;
    float conf_obj   = m1 * d1 * d1 + m2 * d2 * d2;
    float conf_noobj = LAMBDA_NOOBJ * mn * (p4 * p4 + p9 * p9);

    float cls = 0.0f;
#pragma unroll
    for (int k = 5; k < 15; ++k) {      // channels 10..29
        float2 pc = P2[k], tc = T2[k];
        float ex = pc.x - tc.x, ey = pc.y - tc.y;
        cls += ex * ex + ey * ey;
    }
    cls *= mo;

    return LAMBDA_COORD * (loss_xy + loss_wh) + conf_obj + conf_noobj + cls;
}

__global__ void yolo_loss_partial(const float* __restrict__ pred,
                                  const float* __restrict__ targ,
                                  float* __restrict__ partials,
                                  int ncells) {
    int tid    = blockIdx.x * blockDim.x + threadIdx.x;
    int stride = gridDim.x * blockDim.x;

    float acc = 0.0f;
    for (int cell = tid; cell < ncells; cell += stride) {
        int sc   = cell % (S_GRID * S_GRID);
        float gy = (float)(sc / S_GRID);
        float gx = (float)(sc % S_GRID);
        const float* P = pred + (size_t)cell * NCH;
        const float* T = targ + (size_t)cell * NCH;
        int nxt = cell + stride;
        if (nxt < ncells) {           // gfx1250: lowers to global_prefetch_b8
            __builtin_prefetch(pred + (size_t)nxt * NCH, 0, 0);
            __builtin_prefetch(targ + (size_t)nxt * NCH, 0, 0);
        }
        acc += cell_loss(P, T, gx, gy);
    }

    // wave32 reduction
#pragma unroll
    for (int off = 16; off; off >>= 1) acc += __shfl_down(acc, off, 32);

    __shared__ float wsum[8];
    int lane = threadIdx.x & 31;
    int wave = threadIdx.x >> 5;
    if (lane == 0) wsum[wave] = acc;
    __syncthreads();
    if (wave == 0) {
        float v = (lane < (int)(blockDim.x >> 5)) ? wsum[lane] : 0.0f;
#pragma unroll
        for (int off = 4; off; off >>= 1) v += __shfl_down(v, off, 32);
        if (lane == 0) partials[blockIdx.x] = v;
    }
}

// Final reduction: one wave32. Lane sums of the block partials are folded with
// one V_WMMA_F32_16X16X4_F32 (A = ones 16x4, lane values placed in B's first
// VGPR -> every row of D is the column-sum vector; the wave-sum of D's VGPR0
// counts the total exactly twice since rows 0 and 8 are identical).
__global__ void yolo_loss_final(const float* __restrict__ partials,
                                int n, float* __restrict__ out) {
    float v = 0.0f;
    for (int i = (int)threadIdx.x; i < n; i += 32) v += partials[i];

    v2f a; a[0] = 1.0f; a[1] = 1.0f;   // A: 16x4 all-ones
    v2f b; b[0] = v;    b[1] = 0.0f;   // 32 distinct B cells carry the lane sums
    v8f c = {};
    c = __builtin_amdgcn_wmma_f32_16x16x4_f32(
            /*neg_a=*/false, a, /*neg_b=*/false, b,
            /*c_mod=*/(short)0, c, /*reuse_a=*/false, /*reuse_b=*/false);

    float s = c[0];
#pragma unroll
    for (int off = 16; off; off >>= 1) s += __shfl_down(s, off, 32);
    if (threadIdx.x == 0) out[0] = 0.5f * s;
}

extern "C" void kernel_launch(void* const* d_in, const int* in_sizes, int n_in,
                              void* d_out, int out_size, void* d_ws, size_t ws_size,
                              hipStream_t stream) {
    const float* pred = (const float*)d_in[0];
    const float* targ = (const float*)d_in[1];
    int ncells = in_sizes[0] / NCH;          // 8192*7*7 = 401408

    float* partials = (float*)d_ws;          // 1024 floats of scratch
    const int NB = 1024;
    const int NT = 256;                      // 8 wave32 per block

    yolo_loss_partial<<<NB, NT, 0, stream>>>(pred, targ, partials, ncells);
    yolo_loss_final<<<1, 32, 0, stream>>>(partials, NB, (float*)d_out);
}